// IMMLoss_72430328480133
// MI455X (gfx1250) — compile-verified
//
#include <hip/hip_runtime.h>
#include <math.h>
#include <stdint.h>

typedef __attribute__((ext_vector_type(2))) float v2f;
typedef __attribute__((ext_vector_type(8))) float v8f;

#define USE_ASYNC 1

#define B_DIM 32
#define M_DIM 4
#define D_DIM 196608
#define CHUNKS 48
#define CHUNK_D (D_DIM / CHUNKS)            // 4096
#define TILE_D 512
#define TILES (CHUNK_D / TILE_D)            // 8 (even -> last compute uses buf1)
#define LDS_STRIDE 516                      // 512 + 4 pad -> conflict-free 16-row reads
#define NROWS 16
#define NPAIRS (B_DIM / 2)
#define BUF_FLOATS (NROWS * LDS_STRIDE)     // one buffer, 33024 B

static __device__ __forceinline__ void wait_asynccnt0() {
#if USE_ASYNC
#if defined(__has_builtin) && __has_builtin(__builtin_amdgcn_s_wait_asynccnt)
  __builtin_amdgcn_s_wait_asynccnt(0);
#else
  asm volatile("s_wait_asynccnt 0x0" ::: "memory");
#endif
#endif
}

// ---------------------------------------------------------------------------
// Kernel 1: streaming Gram accumulation with V_WMMA_F32_16X16X4_F32.
// grid = (CHUNKS, NPAIRS), block = 256 (8 waves).
// Rows 0..7  = [st0..st3, sr0..sr3] of b = 2*pair
// Rows 8..15 = [st0..st3, sr0..sr3] of b = 2*pair+1
// Double-buffered LDS; next tile streamed with global_load_async_to_lds_b128
// while the current tile feeds a 16-deep WMMA accumulation chain per wave.
// ---------------------------------------------------------------------------
__global__ __launch_bounds__(256) void immloss_gram_wmma(
    const float* __restrict__ f_st,
    const float* __restrict__ f_sr,
    float* __restrict__ partial) {
  __shared__ float lds[2 * BUF_FLOATS];

  const int t    = threadIdx.x;
  const int wave = t >> 5;
  const int lane = t & 31;
  const int chunk = blockIdx.x;
  const int bpair = blockIdx.y;
  const int d0_chunk = chunk * CHUNK_D;

  v8f acc = {};

  // cooperative-staging coordinates: thread t owns float4 columns col4 of
  // rows rbase, rbase+2, ..., rbase+14  (8 rows, 16 B each per tile)
  const int col4  = (t & 127) * 4;
  const int rbase = t >> 7;

  const float* gp[8];     // per-row global cursor (starts at tile 0)
  uint32_t     loff[8];   // per-row LDS byte offset within buffer 0
#pragma unroll
  for (int i = 0; i < 8; ++i) {
    const int row = rbase + 2 * i;
    const int fr  = row & 7;
    const int b   = bpair * 2 + (row >> 3);
    const float* src = (fr < 4) ? f_st : f_sr;
    gp[i]   = src + (size_t)(b * M_DIM + (fr & 3)) * D_DIM + d0_chunk + col4;
    loff[i] = (uint32_t)(uintptr_t)&lds[row * LDS_STRIDE + col4];
  }

  // WMMA A-fragment read coordinates (lane L<16: M=L, k=0,1; L>=16: M=L-16, k=2,3)
  const int frag_off = (lane & 15) * LDS_STRIDE + ((lane >> 4) << 1) + wave * 64;

  // ---- stage a tile into buffer `buf` and advance global cursors ----
  auto stage = [&](int buf) {
    const uint32_t boff = (uint32_t)buf * (BUF_FLOATS * 4u);
#if USE_ASYNC
#pragma unroll
    for (int i = 0; i < 8; ++i) {
      asm volatile("global_load_async_to_lds_b128 %0, %1, off"
                   :: "v"(loff[i] + boff), "v"(gp[i])
                   : "memory");
      gp[i] += TILE_D;
    }
#else
    float4 v[8];
#pragma unroll
    for (int i = 0; i < 8; ++i) {           // batched: 8 loads in flight
      v[i] = *(const float4*)gp[i];
      gp[i] += TILE_D;
    }
#pragma unroll
    for (int i = 0; i < 8; ++i)
      *(float4*)((char*)lds + ((size_t)buf * (BUF_FLOATS * 4u)) +
                 ((rbase + 2 * i) * LDS_STRIDE + col4) * 4u) = v[i];
#endif
  };

  // ---- software pipeline: single barrier per tile ----
  stage(0);                                  // prologue: tile 0 -> buf 0
  for (int tile = 0; tile < TILES; ++tile) {
    const int cur = tile & 1;
    wait_asynccnt0();                        // my writes for tile `tile` landed
    __syncthreads();                         // everyone's landed; prev compute done
    if (tile + 1 < TILES)
      stage(cur ^ 1);                        // overlap next tile with WMMAs

    const float* rp = &lds[cur * BUF_FLOATS + frag_off];
#pragma unroll
    for (int s = 0; s < 16; ++s) {
      v2f a;
      a.x = rp[4 * s];
      a.y = rp[4 * s + 1];
      // B = A^T has the identical register image under the f32 16x16x4
      // layout, so pass the same fragment twice: C += A * A^T.
      acc = __builtin_amdgcn_wmma_f32_16x16x4_f32(
          false, a, false, a, (short)0, acc, false, false);
    }
  }

  // ---- cross-wave reduction of the 16x16 accumulator ----
  // Last compute read buf1; scratch below lives in buf0 -> disjoint, no race.
#pragma unroll
  for (int v = 0; v < 8; ++v)
    lds[wave * 256 + v * 32 + lane] = acc[v];
  __syncthreads();

  float sum = 0.f;
#pragma unroll
  for (int w = 0; w < 8; ++w) sum += lds[w * 256 + t];
  // element t encodes (vgpr v = t/32, lane l = t%32) of the C layout
  partial[((size_t)bpair * CHUNKS + chunk) * 256 + t] = sum;
}

// ---------------------------------------------------------------------------
// Kernel 2: reduce D-chunk partials. grid = NPAIRS, block = 256.
// ---------------------------------------------------------------------------
__global__ void immloss_reduce_chunks(const float* __restrict__ partial,
                                      float* __restrict__ gram) {
  const int bp = blockIdx.x;
  const int t  = threadIdx.x;
  float s = 0.f;
  for (int c = 0; c < CHUNKS; ++c)
    s += partial[((size_t)bp * CHUNKS + c) * 256 + t];
  gram[bp * 256 + t] = s;
}

// ---------------------------------------------------------------------------
// Kernel 3: epilogue. 1 block, 32 threads (one per batch group).
// ---------------------------------------------------------------------------
__global__ void immloss_final(const float* __restrict__ gram,
                              const float* __restrict__ wt,
                              const float* __restrict__ wtout,
                              float* __restrict__ out) {
  const int b  = threadIdx.x;        // 0..31
  const int bp = b >> 1;
  const int hb = b & 1;
  const float* G = gram + bp * 256;

  // G entry (M,N) of C-layout: M<8 -> v=M, l=N ; M>=8 -> v=M-8, l=N+16
  auto g = [&](int i, int j) -> float {
    const int M = hb * 8 + i;
    const int N = hb * 8 + j;
    const int v = M & 7;
    const int l = N + ((M >= 8) ? 16 : 0);
    return G[v * 32 + l];
  };

  const float w    = wt[b];
  const float invD = 1.0f / (float)D_DIM;  // sigma = 1.0

  auto kmean = [&](int ro, int co) -> float {
    float acc = 0.f;
    for (int i = 0; i < 4; ++i)
      for (int j = 0; j < 4; ++j) {
        const float nii = g(ro + i, ro + i);
        const float njj = g(co + j, co + j);
        const float gij = g(ro + i, co + j);
        float d2 = fmaxf(nii + njj - 2.0f * gij, 1e-12f);
        float e  = -sqrtf(d2) * invD * w;
        e = fminf(fmaxf(e, -1.0e6f), 0.0f);
        acc += expf(e);
      }
    return acc * (1.0f / 16.0f);
  };

  const float loss_b = wtout[b] * (kmean(0, 0) + kmean(4, 4) - 2.0f * kmean(0, 4));

  __shared__ float red[32];
  red[b] = loss_b;
  __syncthreads();
  if (b == 0) {
    float s = 0.f;
    for (int i = 0; i < 32; ++i) s += red[i];
    out[0] = s * (1.0f / (float)B_DIM);
  }
}

// ---------------------------------------------------------------------------
extern "C" void kernel_launch(void* const* d_in, const int* in_sizes, int n_in,
                              void* d_out, int out_size, void* d_ws, size_t ws_size,
                              hipStream_t stream) {
  const float* f_st  = (const float*)d_in[0];
  const float* f_sr  = (const float*)d_in[1];
  const float* wt    = (const float*)d_in[2];
  const float* wtout = (const float*)d_in[3];
  float* out = (float*)d_out;

  float* partial = (float*)d_ws;                            // [16][48][256]
  float* gram    = partial + (size_t)NPAIRS * CHUNKS * 256; // [16][256]

  dim3 grid1(CHUNKS, NPAIRS);
  immloss_gram_wmma<<<grid1, 256, 0, stream>>>(f_st, f_sr, partial);
  immloss_reduce_chunks<<<NPAIRS, 256, 0, stream>>>(partial, gram);
  immloss_final<<<1, 32, 0, stream>>>(gram, wt, wtout, out);
  (void)in_sizes; (void)n_in; (void)out_size; (void)ws_size;
}